// HeteroGNN_27702539059750
// MI455X (gfx1250) — compile-verified
//
#include <hip/hip_runtime.h>

#define DCH 64  // feature dim

typedef __attribute__((ext_vector_type(2))) float v2f;
typedef __attribute__((ext_vector_type(8))) float v8f;

// Monotone order-preserving float<->u32 mapping for atomicMax on floats.
__device__ __forceinline__ unsigned flip_f32(float f) {
    unsigned u = __float_as_uint(f);
    return (u & 0x80000000u) ? ~u : (u | 0x80000000u);
}
__device__ __forceinline__ float unflip_f32(unsigned u) {
    return (u & 0x80000000u) ? __uint_as_float(u & 0x7FFFFFFFu)
                             : __uint_as_float(~u);
}

// ---------------------------------------------------------------------------
// Phase 1: col_max[c] = max_r (x[r] . y[c]) / 8   via V_WMMA_F32_16X16X4_F32
// One wave per block (EXEC all-1s required for WMMA). Each block owns a
// 16-column tile and a 256-row slice; B tile (y) lives in 32 VGPRs for the
// whole row loop. Each 16x16 S-tile = 16 chained K=4 WMMAs.
// ---------------------------------------------------------------------------
__global__ __launch_bounds__(32) void colmax_kernel(
    const float* __restrict__ x, const float* __restrict__ y,
    unsigned* __restrict__ colmax_u, int rowsPerBlock) {
    const int lane  = threadIdx.x & 31;
    const int n     = lane & 15;          // col (for B) / row (for A) in tile
    const int khalf = (lane >> 4) * 2;    // lanes 0-15 -> K0,K1 ; 16-31 -> K2,K3
    const int c0    = blockIdx.x * 16;
    const int r0    = blockIdx.y * rowsPerBlock;

    // Preload B tile: B[k][n] = y[c0+n][k], 16 K-steps of 4.
    v2f b[16];
    #pragma unroll
    for (int s = 0; s < 16; ++s) {
        const float* yp = y + (size_t)(c0 + n) * DCH + s * 4 + khalf;
        b[s] = *(const v2f*)yp;
    }

    float runmax = -3.402823466e38f;
    for (int t = 0; t < rowsPerBlock; t += 16) {
        const int r = r0 + t + n;
        v2f a[16];
        #pragma unroll
        for (int s = 0; s < 16; ++s) {
            const float* xp = x + (size_t)r * DCH + s * 4 + khalf;
            a[s] = *(const v2f*)xp;
        }
        v8f acc = {};
        #pragma unroll
        for (int s = 0; s < 16; ++s) {
            // (neg_a, A, neg_b, B, c_mod, C, reuse_a, reuse_b)
            acc = __builtin_amdgcn_wmma_f32_16x16x4_f32(
                false, a[s], false, b[s], (short)0, acc, false, false);
        }
        // Max over the 8 rows this lane holds (fixed column n).
        float m = acc[0];
        #pragma unroll
        for (int i = 1; i < 8; ++i) m = fmaxf(m, acc[i]);
        // Merge rows held by the partner half-wave (same column n).
        m = fmaxf(m, __shfl_xor(m, 16, 32));
        runmax = fmaxf(runmax, m);
    }
    if (lane < 16) {
        atomicMax(colmax_u + c0 + lane, flip_f32(runmax * 0.125f));
    }
}

// ---------------------------------------------------------------------------
// Phase 2: per-edge score. One wave per edge: coalesced 64-wide dot,
// shuffle-reduce, exp, unnormalized p to ws, atomicAdd into denom[c].
// ---------------------------------------------------------------------------
__global__ __launch_bounds__(256) void edge_score_kernel(
    const float* __restrict__ x, const float* __restrict__ y,
    const float* __restrict__ edge_vals, const int* __restrict__ edge_rows,
    const int* __restrict__ edge_cols, const unsigned* __restrict__ colmax_u,
    float* __restrict__ p, float* __restrict__ denom, int E) {
    const int lane = threadIdx.x & 31;
    const int e    = blockIdx.x * 8 + (threadIdx.x >> 5);
    if (e >= E) return;
    const int r = edge_rows[e];
    const int c = edge_cols[e];
    const float* xr = x + (size_t)r * DCH;
    const float* yc = y + (size_t)c * DCH;
    float s = xr[lane] * yc[lane] + xr[lane + 32] * yc[lane + 32];
    #pragma unroll
    for (int off = 16; off >= 1; off >>= 1) s += __shfl_xor(s, off, 32);
    const float cm = unflip_f32(colmax_u[c]);
    const float pe = expf((s * 0.125f - cm) * edge_vals[e]);
    if (lane == 0) {
        p[e] = pe;
        atomicAdd(denom + c, pe);
    }
}

// ---------------------------------------------------------------------------
// Phase 3: out[c] += (p_e / denom[c]) * hidden[r].  Wave per edge,
// 64 f32 atomics into the L2-resident [N,64] output (pre-seeded with parent).
// ---------------------------------------------------------------------------
__global__ __launch_bounds__(256) void aggregate_kernel(
    const float* __restrict__ hidden, const float* __restrict__ p,
    const float* __restrict__ denom, const int* __restrict__ edge_rows,
    const int* __restrict__ edge_cols, float* __restrict__ out, int E) {
    const int lane = threadIdx.x & 31;
    const int e    = blockIdx.x * 8 + (threadIdx.x >> 5);
    if (e >= E) return;
    const int r = edge_rows[e];
    const int c = edge_cols[e];
    float dn = denom[c];
    dn = (dn == 0.0f) ? 1.0f : dn;
    const float w = p[e] / dn;
    const float* hr = hidden + (size_t)r * DCH;
    float* oc = out + (size_t)c * DCH;
    atomicAdd(oc + lane, w * hr[lane]);
    atomicAdd(oc + lane + 32, w * hr[lane + 32]);
}

__global__ void init_kernel(unsigned* __restrict__ colmax_u,
                            float* __restrict__ denom, int N) {
    int i = blockIdx.x * blockDim.x + threadIdx.x;
    if (i < N) { colmax_u[i] = 0u; denom[i] = 0.0f; }
}

__global__ void copy_kernel(const float* __restrict__ src,
                            float* __restrict__ dst, int n) {
    int i = blockIdx.x * blockDim.x + threadIdx.x;
    if (i < n) dst[i] = src[i];
}

extern "C" void kernel_launch(void* const* d_in, const int* in_sizes, int n_in,
                              void* d_out, int out_size, void* d_ws, size_t ws_size,
                              hipStream_t stream) {
    const float* x      = (const float*)d_in[0];
    const float* y      = (const float*)d_in[1];
    const float* hidden = (const float*)d_in[2];
    const float* parent = (const float*)d_in[3];
    const float* evals  = (const float*)d_in[4];
    const int*   erows  = (const int*)d_in[5];
    const int*   ecols  = (const int*)d_in[6];
    float* out = (float*)d_out;

    const int N = in_sizes[1] / DCH;   // y is [N, D]
    const int E = in_sizes[4];

    // Workspace layout: [u32 colmax | f32 denom | f32 p]  (~1.7 MB total)
    unsigned* colmax_u = (unsigned*)d_ws;
    float*    denom    = (float*)(colmax_u + N);
    float*    p        = denom + N;

    init_kernel<<<(N + 255) / 256, 256, 0, stream>>>(colmax_u, denom, N);
    copy_kernel<<<(out_size + 255) / 256, 256, 0, stream>>>(parent, out, out_size);

    const int rowsPerBlock = 256;
    dim3 g1(N / 16, (N + rowsPerBlock - 1) / rowsPerBlock);   // 768 x 48
    colmax_kernel<<<g1, 32, 0, stream>>>(x, y, colmax_u, rowsPerBlock);

    const int edgeBlocks = (E + 7) / 8;   // 8 waves (edges) per 256-thread block
    edge_score_kernel<<<edgeBlocks, 256, 0, stream>>>(x, y, evals, erows, ecols,
                                                      colmax_u, p, denom, E);
    aggregate_kernel<<<edgeBlocks, 256, 0, stream>>>(hidden, p, denom, erows,
                                                     ecols, out, E);
}